// AttentionMechanism_12635793785061
// MI455X (gfx1250) — compile-verified
//
#include <hip/hip_runtime.h>
#include <hip/hip_bf16.h>

// Problem constants (from reference)
#define BB   2048
#define HH   512
#define KK   10
#define AA   80
#define LL   1024
#define IND  595   // A + 3 + H
#define NP   30    // 3*K output params
#define KPAD 600   // padded K extent (>= 149*4 = 596)
#define APAD 602   // LDS row stride (even, bank-conflict-free for 16-row column reads)

typedef float v2f __attribute__((ext_vector_type(2)));
typedef float v8f __attribute__((ext_vector_type(8)));
typedef float f4  __attribute__((ext_vector_type(4)));

// Gather one element of concatenated attn_in = [prev_window(80) | inputs(3) | hidden(512)]
// Only used in the one-time LDS staging loop (branches harmless there).
__device__ __forceinline__ float attn_at(const float* __restrict__ pw,
                                         const float* __restrict__ ins,
                                         const float* __restrict__ hs,
                                         int row, int k) {
    if (k < AA)      return pw[row * AA + k];
    if (k < AA + 3)  return ins[row * 3 + (k - AA)];
    if (k < IND)     return hs[row * HH + (k - AA - 3)];
    return 0.0f;
}

// Branchless W read: clamp address so the load is always valid, select 0 after.
// Lowers to unconditional global_load_b32 + v_cndmask (no exec-mask branches).
__device__ __forceinline__ float w_sel(const float* __restrict__ W, int n, int k) {
    const int nn = n < NP ? n : (NP - 1);
    const int kk = k < IND ? k : (IND - 1);
    const float v = W[nn * IND + kk];
    return (n < NP && k < IND) ? v : 0.0f;
}

// ---------------------------------------------------------------------------
// Kernel 1: params = softplus(attn_in @ W^T + b) via V_WMMA_F32_16X16X4_F32.
// One block = one 16-row batch tile, N padded 30->32 (two 16x16 WMMA tiles).
// attn_in tile staged in LDS (padded stride); K loop (149 steps of 4) split
// across 8 waves; partial C tiles reduced through LDS; epilogue applies
// softplus / beta clamp / kappa update.
// ---------------------------------------------------------------------------
__global__ __launch_bounds__(256)
void params_wmma_kernel(const float* __restrict__ hs,
                        const float* __restrict__ pw,
                        const float* __restrict__ pk,
                        const float* __restrict__ ins,
                        const float* __restrict__ W,
                        const float* __restrict__ bias,
                        float* __restrict__ wsA,    // alpha  (B,10)
                        float* __restrict__ wsB,    // beta   (B,10) clamped
                        float* __restrict__ wsK,    // kappa  (B,10)
                        float* __restrict__ kapOut) // d_out kappa section
{
    __shared__ __align__(16) float sA[16 * APAD];    // 38.5 KB
    __shared__ float s_tile[8][16 * 32];             // 16 KB

    const int tid  = threadIdx.x;
    const int wave = tid >> 5;
    const int lane = tid & 31;
    const int m    = lane & 15;   // A row / D row-half
    const int g    = lane >> 4;   // k-pair selector
    const int n    = lane & 15;   // B/D column within tile
    const int b0   = blockIdx.x * 16;

    // One-time cooperative stage of the 16 x KPAD attn_in tile into LDS.
    for (int r = wave; r < 16; r += 8) {
        const int grow = b0 + r;
        for (int k = lane; k < KPAD; k += 32)
            sA[r * APAD + k] = attn_at(pw, ins, hs, grow, k);
    }
    __syncthreads();

    v8f c0 = {};  // columns 0..15
    v8f c1 = {};  // columns 16..31

    // 149 K-steps of 4, interleaved across the 8 waves. Straight-line body:
    // 1 ds_load_b64 + 4 global_load_b32 + 2 v_wmma.
    for (int s = wave; s < 149; s += 8) {
        const int ka = s * 4 + 2 * g;                 // even -> b64-aligned
        const v2f a = *(const v2f*)(&sA[m * APAD + ka]);
        v2f bt0, bt1;
        bt0.x = w_sel(W, n,      ka);
        bt0.y = w_sel(W, n,      ka + 1);
        bt1.x = w_sel(W, n + 16, ka);
        bt1.y = w_sel(W, n + 16, ka + 1);
        c0 = __builtin_amdgcn_wmma_f32_16x16x4_f32(false, a, false, bt0,
                                                   (short)0, c0, false, false);
        c1 = __builtin_amdgcn_wmma_f32_16x16x4_f32(false, a, false, bt1,
                                                   (short)0, c1, false, false);
    }

    // Spill per-wave D tiles. D layout: col = lane&15, row = r + 8*(lane>>4).
    #pragma unroll
    for (int r = 0; r < 8; ++r) {
        const int mm = r + 8 * g;
        s_tile[wave][mm * 32 + n]      = c0[r];
        s_tile[wave][mm * 32 + 16 + n] = c1[r];
    }
    __syncthreads();

    // Cross-wave reduction + epilogue (512 elements, 256 threads -> 2 each)
    for (int idx = tid; idx < 512; idx += 256) {
        float sum = 0.0f;
        #pragma unroll
        for (int w = 0; w < 8; ++w) sum += s_tile[w][idx];
        const int mm = idx >> 5;
        const int j  = idx & 31;
        if (j < NP) {
            const int gb = b0 + mm;
            const float x  = sum + bias[j];
            // numerically stable softplus = logaddexp(x, 0)
            const float sp = fmaxf(x, 0.0f) + log1pf(expf(-fabsf(x)));
            if (j < KK) {
                wsA[gb * KK + j] = sp;                          // alpha
            } else if (j < 2 * KK) {
                wsB[gb * KK + (j - KK)] = fmaxf(sp, 0.01f);     // beta
            } else {
                const int jk = j - 2 * KK;
                const float kap = pk[gb * KK + jk] + sp * 0.04f; // /25
                wsK[gb * KK + jk]    = kap;
                kapOut[gb * KK + jk] = kap;
            }
        }
    }
}

// ---------------------------------------------------------------------------
// Kernel 2: phi (Gaussian mixture, native v_exp_f32) + context (bandwidth
// bound stream of char_seq).  One block per batch row.  Rows l >= len are
// never read from memory (mask makes them zero) -> ~2x traffic saving.
// ---------------------------------------------------------------------------
__global__ __launch_bounds__(320)
void phi_context_kernel(const float* __restrict__ cs,
                        const int*   __restrict__ lens,
                        const float* __restrict__ wsA,
                        const float* __restrict__ wsB,
                        const float* __restrict__ wsK,
                        float* __restrict__ ctxOut,
                        float* __restrict__ phiOut)
{
    __shared__ float s_par[32];
    __shared__ float s_phi[LL];
    __shared__ f4    s_red[320];

    const int b   = blockIdx.x;
    const int tid = threadIdx.x;

    if (tid < NP) {
        s_par[tid] = (tid < KK)     ? wsA[b * KK + tid]
                   : (tid < 2 * KK) ? wsB[b * KK + (tid - KK)]
                                    : wsK[b * KK + (tid - 2 * KK)];
    }
    __syncthreads();

    float al[KK], iv[KK], kp[KK];
    #pragma unroll
    for (int k = 0; k < KK; ++k) {
        al[k] = s_par[k];
        const float be = s_par[KK + k];
        iv[k] = 0.72134752f / (be * be);     // log2(e) / (2*beta^2)
        kp[k] = s_par[2 * KK + k];
    }

    const int len = lens[b];
    float* __restrict__ phiRow = phiOut + (size_t)b * LL;

    for (int l = tid; l < LL; l += 320) {
        float ph = 0.0f;
        if (l < len) {
            const float u = (float)l;
            #pragma unroll
            for (int k = 0; k < KK; ++k) {
                const float d = kp[k] - u;
                ph += al[k] * exp2f(-d * d * iv[k]);   // native v_exp_f32
            }
        }
        s_phi[l]  = ph;
        phiRow[l] = ph;   // zeros past len (d_out is poisoned)
    }
    __syncthreads();

    // context: 20 float4 columns x 16 row-groups; stream only l < len
    const int c4 = tid % 20;     // float4 column (80 floats = 20 x float4)
    const int rg = tid / 20;     // row group 0..15
    const f4* __restrict__ cs4 = (const f4*)cs + (size_t)b * LL * 20;

    f4 acc = {};
    for (int l = rg; l < len; l += 16) {
        const float p = s_phi[l];
        const f4 v = __builtin_nontemporal_load(&cs4[(size_t)l * 20 + c4]);
        acc += p * v;
    }
    s_red[rg * 20 + c4] = acc;
    __syncthreads();

    for (int s = 8; s > 0; s >>= 1) {
        if (rg < s) {
            acc += s_red[(rg + s) * 20 + c4];
            s_red[rg * 20 + c4] = acc;
        }
        __syncthreads();
    }
    if (rg == 0) {
        f4* out4 = (f4*)(ctxOut + (size_t)b * AA);
        out4[c4] = acc;
    }
}

extern "C" void kernel_launch(void* const* d_in, const int* in_sizes, int n_in,
                              void* d_out, int out_size, void* d_ws, size_t ws_size,
                              hipStream_t stream) {
    const float* hs   = (const float*)d_in[0];  // hidden_state (B,H)
    const float* pw   = (const float*)d_in[1];  // prev_window  (B,A)
    const float* pk   = (const float*)d_in[2];  // prev_kappa   (B,K)
    const float* ins  = (const float*)d_in[3];  // inputs       (B,3)
    const float* cs   = (const float*)d_in[4];  // char_seq     (B,L,A)
    const float* W    = (const float*)d_in[5];  // W            (3K,IN_DIM)
    const float* bias = (const float*)d_in[6];  // b            (3K,)
    const int*   lens = (const int*)d_in[7];    // char_seq_lengths (B,)

    float* out = (float*)d_out;
    float* ctx = out;                       // (B, 80)
    float* kap = out + (size_t)BB * AA;     // (B, 10)
    float* phi = kap + (size_t)BB * KK;     // (B, 1024)

    float* ws  = (float*)d_ws;
    float* wsA = ws;
    float* wsB = ws + (size_t)BB * KK;
    float* wsK = ws + (size_t)2 * BB * KK;

    params_wmma_kernel<<<BB / 16, 256, 0, stream>>>(hs, pw, pk, ins, W, bias,
                                                    wsA, wsB, wsK, kap);
    phi_context_kernel<<<BB, 320, 0, stream>>>(cs, lens, wsA, wsB, wsK, ctx, phi);
}